// LaplaceNet_10960756540115
// MI455X (gfx1250) — compile-verified
//
#include <hip/hip_runtime.h>
#include <hip/hip_bf16.h>

#define N 128
#define N3 (N * N * N)            // 2097152
#define NQ (64 * 64 * 64)         // 262144 query points
#define MAXV 1000.0f
#define BIGF 3.0e38f

// ---------------- separable 5-tap box conv (zero 'SAME' padding) -------------

__global__ __launch_bounds__(256) void k_conv_x(const int* __restrict__ bt,
                                                const float* __restrict__ bc,
                                                float* __restrict__ T1) {
    int i = blockIdx.x * 256 + threadIdx.x;
    if (i >= N3) return;
    int x = i & (N - 1);
    int base = i - x;
    float s = 0.f;
#pragma unroll
    for (int d = -2; d <= 2; ++d) {
        int xx = x + d;
        if (xx >= 0 && xx < N) {
            int j = base + xx;
            float b = bc[j];
            s += (bt[j] == 1 && b > 0.f) ? b : 0.f;   // objects_bounds
        }
    }
    T1[i] = s;
}

__global__ __launch_bounds__(256) void k_conv_y(const float* __restrict__ T1,
                                                float* __restrict__ T2) {
    int i = blockIdx.x * 256 + threadIdx.x;
    if (i >= N3) return;
    int x = i & (N - 1);
    int y = (i >> 7) & (N - 1);
    int zb = i & ~((N * N) - 1);
    float s = 0.f;
#pragma unroll
    for (int d = -2; d <= 2; ++d) {
        int yy = y + d;
        if (yy >= 0 && yy < N) s += T1[zb + (yy << 7) + x];
    }
    T2[i] = s;
}

// z-pass fused with Cmap / Fix / V-init construction
__global__ __launch_bounds__(256) void k_conv_z_init(const float* __restrict__ T2,
                                                     const int* __restrict__ bt,
                                                     const float* __restrict__ bc,
                                                     const float* __restrict__ C,
                                                     float* __restrict__ Cmap,
                                                     float* __restrict__ Fix,
                                                     float* __restrict__ V0) {
    int i = blockIdx.x * 256 + threadIdx.x;
    if (i >= N3) return;
    int xy = i & (N * N - 1);
    int z = i >> 14;
    float s = 0.f;
#pragma unroll
    for (int d = -2; d <= 2; ++d) {
        int zz = z + d;
        if (zz >= 0 && zz < N) s += T2[(zz << 14) + xy];
    }
    int t = bt[i];
    float b = bc[i];
    float c = C[i];
    float cpos = (c > 0.f ? c : 0.f) + 1.0f;                 // relu(C)+COST_SCALE
    float extra = (t == 0) ? s * (1.0f / 125.0f) : 0.f;      // * (bt==0)
    Cmap[i] = cpos + extra;
    Fix[i]  = (t == 1) ? b : BIGF;                           // Dirichlet-fixed cells
    V0[i]   = (t > 0) ? b : MAXV;                            // V init + boundary inject
}

// ---------------- fused 27-point periodic relaxation sweep -------------------
// tile: 16x16 (x,y) * 8 (z) outputs per 256-thread block; halo 18x18x10 in LDS,
// filled with per-lane async global->LDS copies (CDNA5 ASYNCcnt path).

#define TX 16
#define TY 16
#define TZ 8
#define HX (TX + 2)
#define HY (TY + 2)
#define HZ (TZ + 2)
#define HTOT (HX * HY * HZ)   // 3240

__global__ __launch_bounds__(256) void k_relax(const float* __restrict__ Vin,
                                               const float* __restrict__ Cmap,
                                               const float* __restrict__ Fix,
                                               float* __restrict__ Vout) {
    __shared__ float tile[HZ][HY][HX];

    const int t  = threadIdx.x;
    const int x0 = blockIdx.x * TX;
    const int y0 = blockIdx.y * TY;
    const int z0 = blockIdx.z * TZ;

    // cooperative halo fill: per-lane async copy straight into LDS
    for (int i = t; i < HTOT; i += 256) {
        int lx = i % HX;
        int r  = i / HX;
        int ly = r % HY;
        int lz = r / HY;
        int gx = (x0 + lx - 1) & (N - 1);        // periodic wrap (jnp.roll)
        int gy = (y0 + ly - 1) & (N - 1);
        int gz = (z0 + lz - 1) & (N - 1);
        int gi = (gz << 14) + (gy << 7) + gx;
        unsigned int ldsOff = (unsigned int)(uintptr_t)&tile[lz][ly][lx];
        unsigned long long ga = (unsigned long long)(uintptr_t)(Vin + gi);
        asm volatile("global_load_async_to_lds_b32 %0, %1, off"
                     :: "v"(ldsOff), "v"(ga) : "memory");
    }
    asm volatile("s_wait_asynccnt 0" ::: "memory");
    __syncthreads();

    const int tx = t & 15;
    const int ty = t >> 4;

    // sliding-window z reduction: one 3x3 plane sum per z, reused 3x
    float ps[HZ];
#pragma unroll
    for (int z = 0; z < HZ; ++z) {
        float s = 0.f;
#pragma unroll
        for (int dy = 0; dy < 3; ++dy)
#pragma unroll
            for (int dx = 0; dx < 3; ++dx)
                s += tile[z][ty + dy][tx + dx];
        ps[z] = s;
    }

#pragma unroll
    for (int lz = 0; lz < TZ; ++lz) {
        float s27 = ps[lz] + ps[lz + 1] + ps[lz + 2];
        float ctr = tile[lz + 1][ty + 1][tx + 1];
        float mean = (s27 - ctr) * (1.0f / 26.0f);
        int gi = ((z0 + lz) << 14) + ((y0 + ty) << 7) + (x0 + tx);
        float f = Fix[gi];
        float v = (f < 1.0e30f) ? f : (mean + Cmap[gi]);
        Vout[gi] = v;
    }
}

// ---------------- clamp + boundary override ---------------------------------

__global__ __launch_bounds__(256) void k_out(const float* __restrict__ V,
                                             const int* __restrict__ bt,
                                             const float* __restrict__ bc,
                                             float* __restrict__ Out) {
    int i = blockIdx.x * 256 + threadIdx.x;
    if (i >= N3) return;
    float v = V[i];
    float o = (v >= MAXV) ? 0.95f * MAXV : v;
    o = (bt[i] > 0) ? bc[i] : o;
    Out[i] = o;
}

// ---------------- trilinear sampling of out and on-the-fly gradients --------

__device__ __forceinline__ float fget(const float* o, int z, int y, int x) {
    return o[(z << 14) + (y << 7) + x];
}
__device__ __forceinline__ float jxf(const float* o, int z, int y, int x) {
    return (x > 0) ? fget(o, z, y, x) - fget(o, z, y, x - 1) : 0.f;
}
__device__ __forceinline__ float jyf(const float* o, int z, int y, int x) {
    return (y > 0) ? fget(o, z, y, x) - fget(o, z, y - 1, x) : 0.f;
}
__device__ __forceinline__ float jzf(const float* o, int z, int y, int x) {
    return (z > 0) ? fget(o, z, y, x) - fget(o, z - 1, y, x) : 0.f;
}

__global__ __launch_bounds__(256) void k_sample(const float* __restrict__ out,
                                                const float* __restrict__ xq,
                                                float* __restrict__ outq,
                                                float* __restrict__ J) {
    int q = blockIdx.x * 256 + threadIdx.x;
    if (q >= NQ) return;
    float gx = xq[q * 3 + 0];
    float gy = xq[q * 3 + 1];
    float gz = xq[q * 3 + 2];

    float ix = fminf(fmaxf((gx + 1.0f) * 0.5f * (float)(N - 1), 0.f), (float)(N - 1));
    float iy = fminf(fmaxf((gy + 1.0f) * 0.5f * (float)(N - 1), 0.f), (float)(N - 1));
    float iz = fminf(fmaxf((gz + 1.0f) * 0.5f * (float)(N - 1), 0.f), (float)(N - 1));

    int x0 = (int)floorf(ix), y0 = (int)floorf(iy), z0 = (int)floorf(iz);
    int x1 = min(x0 + 1, N - 1), y1 = min(y0 + 1, N - 1), z1 = min(z0 + 1, N - 1);
    float wx = ix - (float)x0, wy = iy - (float)y0, wz = iz - (float)z0;

    auto tri = [&](auto f) -> float {
        float c000 = f(out, z0, y0, x0), c001 = f(out, z0, y0, x1);
        float c010 = f(out, z0, y1, x0), c011 = f(out, z0, y1, x1);
        float c100 = f(out, z1, y0, x0), c101 = f(out, z1, y0, x1);
        float c110 = f(out, z1, y1, x0), c111 = f(out, z1, y1, x1);
        float c00 = c000 * (1.f - wx) + c001 * wx;
        float c01 = c010 * (1.f - wx) + c011 * wx;
        float c10 = c100 * (1.f - wx) + c101 * wx;
        float c11 = c110 * (1.f - wx) + c111 * wx;
        float c0 = c00 * (1.f - wy) + c01 * wy;
        float c1 = c10 * (1.f - wy) + c11 * wy;
        return c0 * (1.f - wz) + c1 * wz;
    };

    outq[q] = tri([] __device__(const float* o, int z, int y, int x) { return fget(o, z, y, x); });
    J[q * 3 + 0] = tri([] __device__(const float* o, int z, int y, int x) { return jxf(o, z, y, x); });
    J[q * 3 + 1] = tri([] __device__(const float* o, int z, int y, int x) { return jyf(o, z, y, x); });
    J[q * 3 + 2] = tri([] __device__(const float* o, int z, int y, int x) { return jzf(o, z, y, x); });
}

// ---------------- host orchestration ----------------------------------------

extern "C" void kernel_launch(void* const* d_in, const int* in_sizes, int n_in,
                              void* d_out, int out_size, void* d_ws, size_t ws_size,
                              hipStream_t stream) {
    (void)in_sizes; (void)n_in; (void)out_size; (void)ws_size;

    const float* xq = (const float*)d_in[0];   // (64,64,64,3)
    const int*   bt = (const int*)d_in[1];     // (128^3) int32
    const float* bc = (const float*)d_in[2];   // (128^3)
    const float* C  = (const float*)d_in[3];   // (128^3)

    float* out = (float*)d_out;                // [out_q | C | J]
    float* ws  = (float*)d_ws;
    float* Va  = ws;                           // 8 MB each
    float* Vb  = ws + N3;
    float* Cm  = ws + 2 * (size_t)N3;
    float* Fx  = ws + 3 * (size_t)N3;

    const int blocks = N3 / 256;               // 8192

    // cost-map conv (separable, zero-padded) + V init + Fix mask
    k_conv_x<<<blocks, 256, 0, stream>>>(bt, bc, Va);
    k_conv_y<<<blocks, 256, 0, stream>>>(Va, Vb);
    k_conv_z_init<<<blocks, 256, 0, stream>>>(Vb, bt, bc, C, Cm, Fx, Va);

    // 5 Jacobi sweeps, ping-pong (kernel boundary = global sync)
    float* cur = Va;
    float* oth = Vb;
    dim3 rg(N / TX, N / TY, N / TZ);           // (8,8,16)
    for (int it = 0; it < 5; ++it) {
        k_relax<<<rg, 256, 0, stream>>>(cur, Cm, Fx, oth);
        float* tmp = cur; cur = oth; oth = tmp;
    }

    // clamp + boundary override -> Va (oth after odd #sweeps is free)
    k_out<<<blocks, 256, 0, stream>>>(cur, bt, bc, oth);

    // C passthrough into the middle of d_out
    hipMemcpyAsync(out + NQ, C, (size_t)N3 * sizeof(float),
                   hipMemcpyDeviceToDevice, stream);

    // trilinear sample of out + gradients
    k_sample<<<NQ / 256, 256, 0, stream>>>(oth, xq, out, out + NQ + N3);
}